// VectorQuantizer_88021059764281
// MI455X (gfx1250) — compile-verified
//
#include <hip/hip_runtime.h>

typedef __attribute__((ext_vector_type(2))) float v2f;
typedef __attribute__((ext_vector_type(8))) float v8f;
typedef __attribute__((ext_vector_type(4))) int   v4i;

#define D_DIM        256
#define ROWS_PER_WG  128      // 8 waves * 16 rows each
#define KT           32       // codebook columns per LDS tile (2 subtiles/wave)
#define AS_STRIDE    260      // 256 + 4 pad: conflict-free b64 reads
#define ES_STRIDE    260
#define ES_TILE      (KT * ES_STRIDE)

#define HAS_ASYNC_LDS __has_builtin(__builtin_amdgcn_global_load_async_to_lds_b128)

// ---------------------------------------------------------------------------
// Kernel 1: esq[k] = sum_d E[k][d]^2   (one wave per codebook row)
// ---------------------------------------------------------------------------
__global__ __launch_bounds__(256) void vq_esq_kernel(const float* __restrict__ E,
                                                     float* __restrict__ esq, int K) {
    int gwave = (int)(blockIdx.x * 256 + threadIdx.x) >> 5;
    int lane  = threadIdx.x & 31;
    if (gwave >= K) return;
    const float* row = E + (size_t)gwave * D_DIM;
    float s = 0.f;
#pragma unroll
    for (int i = 0; i < 2; ++i) {
        float4 v = *(const float4*)(row + lane * 4 + i * 128);
        s += v.x * v.x + v.y * v.y + v.z * v.z + v.w * v.w;
    }
#pragma unroll
    for (int m = 16; m >= 1; m >>= 1) s += __shfl_xor(s, m, 32);
    if (lane == 0) esq[gwave] = s;
}

// ---------------------------------------------------------------------------
// Stage one KT x 256 codebook tile into LDS.
// Async path: GLOBAL_LOAD_ASYNC_TO_LDS_B128 (ASYNCcnt), no VGPR staging.
// ---------------------------------------------------------------------------
__device__ __forceinline__ void stage_tile(const float* __restrict__ E, int k0,
                                           float* buf, int tid) {
#pragma unroll
    for (int j = 0; j < (KT * (D_DIM / 4)) / 256; ++j) {   // 8 x b128 per thread
        int idx = tid + j * 256;
        int c   = idx >> 6;
        int f4  = idx & 63;
        const float* gp = E + (size_t)(k0 + c) * D_DIM + f4 * 4;
        float*       lp = &buf[c * ES_STRIDE + f4 * 4];
#if HAS_ASYNC_LDS
        __builtin_amdgcn_global_load_async_to_lds_b128(
            (__attribute__((address_space(1))) v4i*)
                (__attribute__((address_space(1))) void*)gp,
            (__attribute__((address_space(3))) v4i*)
                (__attribute__((address_space(3))) void*)lp,
            0, 0);
#else
        *(float4*)lp = *(const float4*)gp;
#endif
    }
}

__device__ __forceinline__ void wait_stage() {
#if HAS_ASYNC_LDS
    asm volatile("s_wait_asynccnt 0x0" ::: "memory");
#endif
}

// ---------------------------------------------------------------------------
// Kernel 2: fused GEMM (WMMA f32 16x16x4) + argmin epilogue.
//   score[n,k] = esq[k] - 2*<x_n, e_k>;  out[n] = argmin_k score
// ---------------------------------------------------------------------------
__global__ __launch_bounds__(256) void vq_argmin_kernel(const float* __restrict__ X,
                                                        const float* __restrict__ E,
                                                        const float* __restrict__ esq,
                                                        int* __restrict__ out, int K) {
    __shared__ float As[ROWS_PER_WG * AS_STRIDE];  // 133,120 B, resident all kernel
    __shared__ float Es[2 * ES_TILE];              // 66,560 B, double buffered

    const int tid  = threadIdx.x;
    const int lane = tid & 31;
    const int wave = tid >> 5;       // 0..7
    const int half = lane >> 4;      // selects K-pair {0,1} vs {2,3} in A/B frags
    const int l15  = lane & 15;
    const int row0 = blockIdx.x * ROWS_PER_WG;

    // Stage the 128x256 X tile into LDS once (reused for all K tiles).
    for (int idx = tid; idx < ROWS_PER_WG * (D_DIM / 4); idx += 256) {
        int r  = idx >> 6;
        int f4 = idx & 63;
        float4 v = *(const float4*)(X + (size_t)(row0 + r) * D_DIM + f4 * 4);
        *(float4*)&As[r * AS_STRIDE + f4 * 4] = v;
    }

    // Preload codebook tile 0 into buffer 0.
    stage_tile(E, 0, &Es[0], tid);
    wait_stage();
    __syncthreads();

    const float* Abase = &As[(wave * 16 + l15) * AS_STRIDE + 2 * half];

    float bestd[8];
    int   besti[8];
#pragma unroll
    for (int r = 0; r < 8; ++r) { bestd[r] = 3.4e38f; besti[r] = 0; }

    const int ntiles = K / KT;
    for (int t = 0; t < ntiles; ++t) {
        const int    k0  = t * KT;
        const float* cur = &Es[(t & 1) * ES_TILE];

        // Kick off next tile's async fill + prefetch tile t+2 into caches.
        if (t + 1 < ntiles) stage_tile(E, k0 + KT, &Es[((t + 1) & 1) * ES_TILE], tid);
        if (k0 + 2 * KT < K)
            __builtin_prefetch(E + (size_t)(k0 + 2 * KT) * D_DIM + tid * 32, 0, 1);

        // Hoist epilogue operands ahead of the compute loop.
        const float ek0 = esq[k0 + l15];
        const float ek1 = esq[k0 + 16 + l15];

        const float* B0 = cur + (l15)      * ES_STRIDE + 2 * half;
        const float* B1 = cur + (16 + l15) * ES_STRIDE + 2 * half;

        // 128 chained V_WMMA_F32_16X16X4_F32: one A frag feeds two column tiles.
        v8f acc0 = {}, acc1 = {};
#pragma unroll 8
        for (int d = 0; d < D_DIM; d += 4) {
            v2f a  = *(const v2f*)(Abase + d);
            v2f b0 = *(const v2f*)(B0 + d);
            v2f b1 = *(const v2f*)(B1 + d);
            acc0 = __builtin_amdgcn_wmma_f32_16x16x4_f32(
                false, a, false, b0, (short)0, acc0, false, false);
            acc1 = __builtin_amdgcn_wmma_f32_16x16x4_f32(
                false, a, false, b1, (short)0, acc1, false, false);
        }

        // Fused argmin epilogue (k ascending within a lane: first-min wins).
#pragma unroll
        for (int r = 0; r < 8; ++r) {
            float d0 = ek0 - 2.0f * acc0[r];
            if (d0 < bestd[r]) { bestd[r] = d0; besti[r] = k0 + l15; }
            float d1 = ek1 - 2.0f * acc1[r];
            if (d1 < bestd[r]) { bestd[r] = d1; besti[r] = k0 + 16 + l15; }
        }

        wait_stage();        // drain ASYNCcnt so next tile's LDS data is valid
        __syncthreads();     // and all waves are done reading `cur`
    }

    // Per-row argmin across the 16 lanes of each half-group.
#pragma unroll
    for (int r = 0; r < 8; ++r) {
        float d = bestd[r];
        int   i = besti[r];
#pragma unroll
        for (int m = 8; m >= 1; m >>= 1) {
            float od = __shfl_xor(d, m, 32);
            int   oi = __shfl_xor(i, m, 32);
            if (od < d || (od == d && oi < i)) { d = od; i = oi; }
        }
        if (l15 == 0) out[row0 + wave * 16 + half * 8 + r] = i;
    }
}

// ---------------------------------------------------------------------------
extern "C" void kernel_launch(void* const* d_in, const int* in_sizes, int n_in,
                              void* d_out, int out_size, void* d_ws, size_t ws_size,
                              hipStream_t stream) {
    const float* X = (const float*)d_in[0];   // [N, 256] f32
    const float* E = (const float*)d_in[1];   // [K, 256] f32
    const int N = in_sizes[0] / D_DIM;
    const int K = in_sizes[1] / D_DIM;

    float* esq = (float*)d_ws;                // K floats of scratch
    int*   out = (int*)d_out;                 // argmin indices (int32)

    vq_esq_kernel<<<(K + 7) / 8, 256, 0, stream>>>(E, esq, K);
    vq_argmin_kernel<<<N / ROWS_PER_WG, 256, 0, stream>>>(X, E, esq, out, K);
}